// ConfidenceGatedSymmetryModule_65283502899642
// MI455X (gfx1250) — compile-verified
//
#include <hip/hip_runtime.h>
#include <hip/hip_bf16.h>

typedef __attribute__((ext_vector_type(2))) float v2f;
typedef __attribute__((ext_vector_type(8))) float v8f;

#define B_ 8
#define N_ 2048
#define T_ 64
#define D_ 256
#define K_ 128
#define NTILES (N_ / 16)   // 128
#define THR2 (0.05f * 0.05f)

// ---------------------------------------------------------------------------
// Kernel 1: one block per (b,t) hypothesis. Computes frac, normal, offset.
// 8 waves/block, each wave owns M-tiles {wave, wave+8, ...} (16 tiles).
// inner[m][n] = |p_n|^2 - 2 * refl_m . p_n   via  V_WMMA_F32_16X16X4_F32
// matched(m)  <=>  min_n inner[m][n] < thr^2 - |refl_m|^2
// ---------------------------------------------------------------------------
__global__ __launch_bounds__(256) void detect_kernel(
    const float* __restrict__ points, const int* __restrict__ sidx,
    float* __restrict__ ws_frac, float* __restrict__ ws_n,
    float* __restrict__ ws_off)
{
    __shared__ float4 spts[N_];     // [x, y, z, |p|^2]
    __shared__ float  splane[8];    // nnx,nny,nnz,offset, nx,ny,nz,nd
    __shared__ int    scount;

    const int bid = blockIdx.x;         // b*T + t
    const int b   = bid >> 6;
    const int tid = threadIdx.x;
    const float* pb = points + (size_t)b * N_ * 3;

    for (int i = tid; i < N_; i += 256) {
        float x = pb[i * 3 + 0], y = pb[i * 3 + 1], z = pb[i * 3 + 2];
        spts[i] = make_float4(x, y, z, x * x + y * y + z * z);
    }
    if (tid == 0) {
        scount = 0;
        int i1 = sidx[bid * 2 + 0];
        int i2 = sidx[bid * 2 + 1];
        float p1x = pb[i1 * 3], p1y = pb[i1 * 3 + 1], p1z = pb[i1 * 3 + 2];
        float p2x = pb[i2 * 3], p2y = pb[i2 * 3 + 1], p2z = pb[i2 * 3 + 2];
        float dx = p2x - p1x, dy = p2y - p1y, dz = p2z - p1z;
        float nd = sqrtf(dx * dx + dy * dy + dz * dz);
        float inv = 1.0f / (nd + 1e-12f);
        float nx = dx * inv, ny = dy * inv, nz = dz * inv;       // stored normal
        float off = 0.5f * (nx * (p1x + p2x) + ny * (p1y + p2y) + nz * (p1z + p2z));
        float nrm = sqrtf(nx * nx + ny * ny + nz * nz);          // re-normalize (faithful)
        float inv2 = 1.0f / (nrm + 1e-12f);
        splane[0] = nx * inv2; splane[1] = ny * inv2; splane[2] = nz * inv2;
        splane[3] = off;
        splane[4] = nx; splane[5] = ny; splane[6] = nz; splane[7] = nd;
    }
    __syncthreads();

    const float nnx = splane[0], nny = splane[1], nnz = splane[2], off = splane[3];
    const int lane = tid & 31;
    const int wave = tid >> 5;
    const int half = lane >> 4;       // 0: lanes 0-15, 1: lanes 16-31
    const int lrow = lane & 15;
    const float2* sp2 = (const float2*)spts;

    int myCount = 0;
    for (int mt = wave; mt < NTILES; mt += 8) {
        // Build A tile: row r = mt*16 + lrow. A[m] = {-2rx, -2ry, -2rz, 1}
        int r = mt * 16 + lrow;
        float4 p = spts[r];
        float sd = p.x * nnx + p.y * nny + p.z * nnz - off;
        float rx = p.x - 2.0f * sd * nnx;
        float ry = p.y - 2.0f * sd * nny;
        float rz = p.z - 2.0f * sd * nnz;
        float rsq = rx * rx + ry * ry + rz * rz;
        v2f a;
        a.x = half ? (-2.0f * rz) : (-2.0f * rx);   // K=2 | K=0
        a.y = half ? 1.0f         : (-2.0f * ry);   // K=3 | K=1

        v8f rm;
        #pragma unroll
        for (int k = 0; k < 8; ++k) rm[k] = 3.0e38f;
        v8f czero = {};

        for (int j = 0; j < NTILES; ++j) {
            // B tile: col c. lanes 0-15: (x,y)=K0,K1 ; lanes 16-31: (z,|p|^2)=K2,K3
            int c = j * 16 + lrow;
            float2 bb = sp2[c * 2 + half];          // one ds_load_b64, conflict-free
            v2f bv; bv.x = bb.x; bv.y = bb.y;
            v8f cc = __builtin_amdgcn_wmma_f32_16x16x4_f32(
                false, a, false, bv, (short)0, czero, false, false);
            #pragma unroll
            for (int k = 0; k < 8; ++k) rm[k] = fminf(rm[k], cc[k]);
        }

        // Row-min across 16 columns (butterfly stays inside each 16-lane half).
        #pragma unroll
        for (int k = 0; k < 8; ++k) {
            float v = rm[k];
            v = fminf(v, __shfl_xor(v, 1, 32));
            v = fminf(v, __shfl_xor(v, 2, 32));
            v = fminf(v, __shfl_xor(v, 4, 32));
            v = fminf(v, __shfl_xor(v, 8, 32));
            rm[k] = v;   // lanes<16: min of row k ; lanes>=16: min of row k+8
        }
        // Compare vs thr^2 - rsq[row]; count matched rows via ballot bits 0 & 16.
        int cnt = 0;
        #pragma unroll
        for (int k = 0; k < 8; ++k) {
            float rq = __shfl(rsq, k + half * 8, 32);   // rsq of row (k + 8*half)
            bool matched = rm[k] < (THR2 - rq);
            unsigned m = (unsigned)__ballot(matched);
            if (lane == 0) cnt += (m & 1u) + ((m >> 16) & 1u);
        }
        if (lane == 0) myCount += cnt;
    }
    if (lane == 0) atomicAdd(&scount, myCount);
    __syncthreads();

    if (tid == 0) {
        float nd = splane[7];
        ws_frac[bid] = (nd < 1e-8f) ? -1.0f : (float)scount * (1.0f / (float)N_);
        ws_n[bid * 3 + 0] = splane[4];
        ws_n[bid * 3 + 1] = splane[5];
        ws_n[bid * 3 + 2] = splane[6];
        ws_off[bid] = splane[3];
    }
}

// ---------------------------------------------------------------------------
// Kernel 2: per-batch centroid + deterministic sequential best-scan.
// ---------------------------------------------------------------------------
__global__ __launch_bounds__(256) void reduce_kernel(
    const float* __restrict__ points, const float* __restrict__ ws_frac,
    const float* __restrict__ ws_n, const float* __restrict__ ws_off,
    float* __restrict__ out_nrm, float* __restrict__ out_off,
    float* __restrict__ out_cnf, float* __restrict__ ws_sym)
{
    __shared__ float red[256 * 3];
    const int b = blockIdx.x, tid = threadIdx.x;
    const float* pb = points + (size_t)b * N_ * 3;
    float sx = 0.f, sy = 0.f, sz = 0.f;
    for (int i = tid; i < N_; i += 256) {
        sx += pb[i * 3]; sy += pb[i * 3 + 1]; sz += pb[i * 3 + 2];
    }
    red[tid * 3] = sx; red[tid * 3 + 1] = sy; red[tid * 3 + 2] = sz;
    __syncthreads();
    if (tid == 0) {
        float cx = 0.f, cy = 0.f, cz = 0.f;
        for (int i = 0; i < 256; ++i) { cx += red[i * 3]; cy += red[i * 3 + 1]; cz += red[i * 3 + 2]; }
        cx *= (1.0f / N_); cy *= (1.0f / N_); cz *= (1.0f / N_);
        float bn0 = 0.f, bn1 = 1.f, bn2 = 0.f, bo = 0.f, bc = 0.f;
        for (int t = 0; t < T_; ++t) {
            float f = ws_frac[b * T_ + t];
            if (f > bc) {
                bc = f;
                bn0 = ws_n[(b * T_ + t) * 3 + 0];
                bn1 = ws_n[(b * T_ + t) * 3 + 1];
                bn2 = ws_n[(b * T_ + t) * 3 + 2];
                bo  = ws_off[b * T_ + t];
            }
        }
        float sd = bn0 * cx + bn1 * cy + bn2 * cz - bo;
        out_nrm[b * 3 + 0] = bn0; out_nrm[b * 3 + 1] = bn1; out_nrm[b * 3 + 2] = bn2;
        out_off[b] = bo; out_cnf[b] = bc;
        ws_sym[b * 4 + 0] = bn0; ws_sym[b * 4 + 1] = bn1;
        ws_sym[b * 4 + 2] = bn2; ws_sym[b * 4 + 3] = sd;
    }
}

// ---------------------------------------------------------------------------
// Kernel 3: z_enc_aug = [z_enc | sym_feats] @ W^T + bias   (8 x 260 x 256)
// ---------------------------------------------------------------------------
__global__ __launch_bounds__(256) void head_kernel(
    const float* __restrict__ z_enc, const float* __restrict__ W,
    const float* __restrict__ bias, const float* __restrict__ ws_sym,
    float* __restrict__ out_aug)
{
    __shared__ float sz[D_ + 4];
    const int b = blockIdx.x, o = threadIdx.x;
    sz[o] = z_enc[b * D_ + o];
    if (o < 4) sz[D_ + o] = ws_sym[b * 4 + o];
    __syncthreads();
    const float* wr = W + (size_t)o * (D_ + 4);
    float acc = bias[o];
    for (int k = 0; k < D_ + 4; ++k) acc = fmaf(sz[k], wr[k], acc);
    out_aug[b * D_ + o] = acc;
}

// ---------------------------------------------------------------------------
// Kernel 4: z_local passthrough copy (float4, 65536 vectors)
// ---------------------------------------------------------------------------
__global__ __launch_bounds__(256) void copy_kernel(
    const float4* __restrict__ src, float4* __restrict__ dst, int n4)
{
    int i = blockIdx.x * blockDim.x + threadIdx.x;
    if (i < n4) dst[i] = src[i];
}

extern "C" void kernel_launch(void* const* d_in, const int* in_sizes, int n_in,
                              void* d_out, int out_size, void* d_ws, size_t ws_size,
                              hipStream_t stream) {
    const float* points  = (const float*)d_in[0];
    const float* z_enc   = (const float*)d_in[1];
    const float* z_local = (const float*)d_in[2];
    // d_in[3] = proxy_coords (unused by reference outputs)
    const int*   sidx    = (const int*)d_in[4];
    const float* W       = (const float*)d_in[5];
    const float* bias    = (const float*)d_in[6];

    float* ws      = (float*)d_ws;
    float* ws_frac = ws;              // 512
    float* ws_n    = ws + 512;        // 1536
    float* ws_off  = ws + 2048;       // 512
    float* ws_sym  = ws + 2560;       // 32

    float* out     = (float*)d_out;
    float* out_aug = out;                       // 8*256
    float* out_zl  = out + 2048;                // 8*128*256
    float* out_nrm = out + 2048 + 262144;       // 8*3
    float* out_off = out_nrm + 24;              // 8
    float* out_cnf = out_off + 8;               // 8

    detect_kernel<<<B_ * T_, 256, 0, stream>>>(points, sidx, ws_frac, ws_n, ws_off);
    reduce_kernel<<<B_, 256, 0, stream>>>(points, ws_frac, ws_n, ws_off,
                                          out_nrm, out_off, out_cnf, ws_sym);
    head_kernel<<<B_, 256, 0, stream>>>(z_enc, W, bias, ws_sym, out_aug);
    copy_kernel<<<(65536 + 255) / 256, 256, 0, stream>>>(
        (const float4*)z_local, (float4*)out_zl, 65536);
}